// CTCGreedySearch_19877108646658
// MI455X (gfx1250) — compile-verified
//
#include <hip/hip_runtime.h>
#include <stdint.h>

// Problem constants from the reference setup (V is fixed; T,N derived from in_sizes).
#define VOCAB 1024
#define BLANK (VOCAB - 1)   // (-1 + V) % V

typedef float v2f __attribute__((ext_vector_type(2)));
typedef float v8f __attribute__((ext_vector_type(8)));

// wave32 cross-lane reads via ds_bpermute (index in bytes, bits [6:2] used -> wraps mod 32 lanes)
__device__ __forceinline__ int perm_i(int srcLane, int x) {
  return __builtin_amdgcn_ds_bpermute(srcLane << 2, x);
}
__device__ __forceinline__ float perm_f(int srcLane, float x) {
  return __builtin_bit_cast(float,
           __builtin_amdgcn_ds_bpermute(srcLane << 2, __builtin_bit_cast(int, x)));
}

// ---------------------------------------------------------------------------
// Pass 1: one wave32 per (t,n) row of V=1024 fp32 logits.
//   rowmax[row] = max_v log_softmax = -log(sum exp(x - max(x)))
//   rowarg[row] = argmax_v (first occurrence on ties)
// Bandwidth-bound: 8x global_load_b128 per lane, single read of 256 MB.
// Cross-lane sum of exp-partials done with V_WMMA_F32_16X16X4_F32 (B = ones).
// ---------------------------------------------------------------------------
__global__ void __launch_bounds__(256)
ctc_rowreduce(const float* __restrict__ logits,
              float* __restrict__ rowmax,
              int*   __restrict__ rowarg,
              int nrows) {
  const int lane = threadIdx.x & 31;
  const int row  = (int)((blockIdx.x * blockDim.x + threadIdx.x) >> 5);
  if (row >= nrows) return;  // wave-uniform (one wave == one row)

  const float4* p = (const float4*)logits + (size_t)row * (VOCAB / 4);
  float4 v[8];
#pragma unroll
  for (int i = 0; i < 8; ++i) v[i] = p[i * 32 + lane];   // coalesced 512B/instr

  // lane-local max / argmax (increasing index order -> strict '>' keeps first)
  float m = -__builtin_inff();
  int   am = 0;
#pragma unroll
  for (int i = 0; i < 8; ++i) {
    const int base = (i * 32 + lane) * 4;
    if (v[i].x > m) { m = v[i].x; am = base;     }
    if (v[i].y > m) { m = v[i].y; am = base + 1; }
    if (v[i].z > m) { m = v[i].z; am = base + 2; }
    if (v[i].w > m) { m = v[i].w; am = base + 3; }
  }
  // cross-lane argmax reduce (smaller index wins ties)
#pragma unroll
  for (int off = 16; off; off >>= 1) {
    const float om = perm_f(lane ^ off, m);
    const int   oa = perm_i(lane ^ off, am);
    if (om > m || (om == m && oa < am)) { m = om; am = oa; }
  }

  // lane-local exp-sum against the row max (fp32 throughout)
  float s = 0.0f;
#pragma unroll
  for (int i = 0; i < 8; ++i) {
    s += __expf(v[i].x - m) + __expf(v[i].y - m) +
         __expf(v[i].z - m) + __expf(v[i].w - m);
  }

  // Cross-lane f32 sum on the matrix pipe:
  // A (16x4 f32) layout: lanes 0-15 hold {K0,K1}, lanes 16-31 hold {K2,K3}.
  // With A = {s, 0} per lane and B = all-ones: D[m][n] = s_m + s_{m+16}.
  v2f a; a[0] = s;    a[1] = 0.0f;
  v2f b; b[0] = 1.0f; b[1] = 1.0f;
  v8f c = {};
  c = __builtin_amdgcn_wmma_f32_16x16x4_f32(false, a, false, b, (short)0, c,
                                            false, false);
  // lanes 0-15 hold D rows 0..7, lanes 16-31 hold D rows 8..15
  const float h = c[0] + c[1] + c[2] + c[3] + c[4] + c[5] + c[6] + c[7];
  const float tot = h + perm_f(lane ^ 16, h);   // full sum of 32 partials

  if (lane == 0) {
    rowmax[row] = -__logf(tot);
    rowarg[row] = am;
  }
}

// ---------------------------------------------------------------------------
// Pass 2: one wave32 per batch element n. Sequential CTC collapse over T via
// ballot/popcount stream compaction with a cross-chunk carry.
// masked_scatter semantics: phase A fills paths with raw argmax (the
// "undefined" tail), phase B overwrites the compacted prefix in program order.
// d_out layout (all f32): [max_total N | paths T*N (T,N) | out_lens N]
// ---------------------------------------------------------------------------
__global__ void __launch_bounds__(32)
ctc_collapse(const float* __restrict__ rowmax,
             const int*   __restrict__ rowarg,
             const int*   __restrict__ in_lens,
             float* __restrict__ out,
             int T, int N) {
  const int n    = blockIdx.x;
  const int lane = threadIdx.x;
  const int L    = in_lens[n];

  float* max_total = out;
  float* paths     = out + N;
  float* out_lens  = out + N + (size_t)T * N;

  // Phase A: background fill + masked score accumulation
  float acc = 0.0f;
  for (int t0 = 0; t0 < T; t0 += 32) {
    const int t = t0 + lane;
    const size_t idx = (size_t)t * N + n;
    const int a = rowarg[idx];
    if (t < L) acc += rowmax[idx];
    paths[idx] = (float)a;
  }

  // Phase B: dedup + blank removal + compaction scatter
  int pos_base = 0;
  int carry = -1;  // argmax >= 0, so first element never matches "previous"
  for (int t0 = 0; t0 < T; t0 += 32) {
    const int t = t0 + lane;
    const int a = rowarg[(size_t)t * N + n];
    int prev = perm_i(lane - 1, a);  // lane 0 wraps to 31; overridden by carry
    if (lane == 0) prev = carry;
    const bool keep = (t < L) && (a != BLANK) && (a != prev);
    const unsigned mask = __builtin_amdgcn_ballot_w32(keep);
    if (keep) {
      const int pos = pos_base + __builtin_popcount(mask & ((1u << lane) - 1u));
      paths[(size_t)pos * N + n] = (float)a;   // pos <= t: overwrites phase A
    }
    pos_base += __builtin_popcount(mask);
    carry = perm_i(31, a);  // broadcast last raw argmax of the chunk
  }

  // cross-lane sum of the score
#pragma unroll
  for (int off = 16; off; off >>= 1) acc += perm_f(lane ^ off, acc);

  if (lane == 0) {
    max_total[n] = acc;
    out_lens[n]  = (float)pos_base;
  }
}

// ---------------------------------------------------------------------------
extern "C" void kernel_launch(void* const* d_in, const int* in_sizes, int n_in,
                              void* d_out, int out_size, void* d_ws, size_t ws_size,
                              hipStream_t stream) {
  (void)n_in; (void)out_size; (void)ws_size;

  const float* logits  = (const float*)d_in[0];
  const int*   in_lens = (const int*)d_in[1];
  float*       out     = (float*)d_out;

  const int total = in_sizes[0];          // T * N * V
  const int N     = in_sizes[1];          // 32
  const int nrows = total / VOCAB;        // T * N
  const int T     = nrows / N;            // 2048

  float* rowmax = (float*)d_ws;                                        // nrows f32
  int*   rowarg = (int*)((char*)d_ws + (size_t)nrows * sizeof(float)); // nrows i32

  const int blocks = (nrows + 7) / 8;     // 8 waves (256 threads) per block
  hipLaunchKernelGGL(ctc_rowreduce, dim3(blocks), dim3(256), 0, stream,
                     logits, rowmax, rowarg, nrows);
  hipLaunchKernelGGL(ctc_collapse, dim3(N), dim3(32), 0, stream,
                     rowmax, rowarg, in_lens, out, T, N);
}